// VTraceModel_15857019256859
// MI455X (gfx1250) — compile-verified
//
#include <hip/hip_runtime.h>
#include <stdint.h>

#define T_STEPS   1024
#define B_DIM     1024
#define A_DIM     18
#define TB        (T_STEPS * B_DIM)            // 1,048,576
#define ROWS_TOTAL ((T_STEPS + 1) * B_DIM)     // 1,049,600
#define ROWS_PER_BLOCK 256
#define NBLK_ROWS (ROWS_TOTAL / ROWS_PER_BLOCK) // 4100 (exact)
#define NSEG      16
#define SEGLEN    (T_STEPS / NSEG)             // 64
#define NB        (NSEG * B_DIM)               // 16384
#define DISCOUNT_F 0.99f

// ---------------------------------------------------------------------------
// CDNA5 async global->LDS copy (ASYNCcnt-tracked data mover path).
// Builtin signature (from clang diagnostic): pointee type is int2 (b64);
// global side is addrspace(1), LDS side addrspace(3), then imm offset, cpol.
// ---------------------------------------------------------------------------
typedef __attribute__((ext_vector_type(2))) int v2i;
typedef __attribute__((address_space(1))) v2i* g_v2i_p;
typedef __attribute__((address_space(3))) v2i* l_v2i_p;

__device__ __forceinline__ void async_copy_b64(const void* g, void* l) {
#if __has_builtin(__builtin_amdgcn_global_load_async_to_lds_b64)
  __builtin_amdgcn_global_load_async_to_lds_b64((g_v2i_p)g, (l_v2i_p)l, 0, 0);
#else
  uint32_t lds_off = (uint32_t)(size_t)(__attribute__((address_space(3))) char*)l;
  asm volatile("global_load_async_to_lds_b64 %0, %1, off"
               :: "v"(lds_off), "v"((uint64_t)(size_t)g)
               : "memory");
#endif
}

__device__ __forceinline__ void wait_async0() {
#if __has_builtin(__builtin_amdgcn_s_wait_asynccnt)
  __builtin_amdgcn_s_wait_asynccnt(0);
#else
  asm volatile("s_wait_asynccnt 0x0" ::: "memory");
#endif
}

// ---------------------------------------------------------------------------
// Kernel 1: per-row softmax stats. Each block stages a 256-row x 18-float tile
// of both logit tensors into LDS via async DMA, then computes:
//   entropy (all rows), and for t < T: log pi_target(a), rho = pi_t/pi_b.
// Thread tid owns row (block*256 + tid); its 72B chunk sits at byte offset tid*72 in each LDS tile.
// ---------------------------------------------------------------------------
__global__ __launch_bounds__(256) void k_rows(
    const float* __restrict__ tlog, const float* __restrict__ blog,
    const int* __restrict__ actions,
    float* __restrict__ rho_ws, float* __restrict__ logp_ws,
    float* __restrict__ entP) {
  __shared__ float sT[ROWS_PER_BLOCK * A_DIM];   // 18 KB
  __shared__ float sB[ROWS_PER_BLOCK * A_DIM];   // 18 KB
  __shared__ float sred[ROWS_PER_BLOCK];

  const int tid = threadIdx.x;
  const long long row0 = (long long)blockIdx.x * ROWS_PER_BLOCK;
  const long long base = row0 * A_DIM;           // float index of tile start

  // Stage tiles: each thread copies its own 72-byte row as 9 x b64 async ops.
  {
    const char* gT = (const char*)(tlog + base) + tid * 72;
    const char* gB = (const char*)(blog + base) + tid * 72;
    char* lT = (char*)sT + tid * 72;
    char* lB = (char*)sB + tid * 72;
#pragma unroll
    for (int k = 0; k < 9; ++k) async_copy_b64(gT + k * 8, lT + k * 8);
#pragma unroll
    for (int k = 0; k < 9; ++k) async_copy_b64(gB + k * 8, lB + k * 8);
    wait_async0();           // per-wave: own chunks complete before LDS reads
  }

  const long long r = row0 + tid;
  const int act = actions[r];                    // valid for all (T+1)*B rows
  const float* xr = &sT[tid * A_DIM];
  const float* yr = &sB[tid * A_DIM];

  // target row: max, sum exp, sum x*exp  -> logZ, entropy, logp(action)
  float x[A_DIM];
#pragma unroll
  for (int a = 0; a < A_DIM; ++a) x[a] = xr[a];
  float m = x[0];
#pragma unroll
  for (int a = 1; a < A_DIM; ++a) m = fmaxf(m, x[a]);
  float s = 0.f, dot = 0.f;
#pragma unroll
  for (int a = 0; a < A_DIM; ++a) {
    float e = expf(x[a] - m);
    s += e;
    dot += x[a] * e;
  }
  const float logZ = m + logf(s);
  const float ent = logZ - dot / s;              // H = logZ - E_p[x]
  const float lpt = xr[act] - logZ;              // dynamic index -> LDS read

  // behaviour row: only logZ needed
  float y[A_DIM];
#pragma unroll
  for (int a = 0; a < A_DIM; ++a) y[a] = yr[a];
  float m2 = y[0];
#pragma unroll
  for (int a = 1; a < A_DIM; ++a) m2 = fmaxf(m2, y[a]);
  float s2 = 0.f;
#pragma unroll
  for (int a = 0; a < A_DIM; ++a) s2 += expf(y[a] - m2);
  const float lpb = yr[act] - (m2 + logf(s2));

  if (r < (long long)TB) {                       // t < T rows only
    rho_ws[r] = expf(lpt - lpb);
    logp_ws[r] = lpt;
  }

  // block entropy partial (deterministic tree)
  sred[tid] = ent;
  __syncthreads();
#pragma unroll
  for (int w = 128; w > 0; w >>= 1) {
    if (tid < w) sred[tid] += sred[tid + w];
    __syncthreads();
  }
  if (tid == 0) entP[blockIdx.x] = sred[0];
}

// ---------------------------------------------------------------------------
// Kernel 2a: segmented backward linear recurrence acc = delta + a*acc.
// Per (segment s, column b): reduce to (P = prod a_t, D) with
// acc(segStart) = D + P * acc(segEnd).  64 blocks x 256 threads, coalesced.
// ---------------------------------------------------------------------------
__global__ __launch_bounds__(256) void k_seg(
    const float* __restrict__ rho_ws, const float* __restrict__ vp,
    const float* __restrict__ rw, const int* __restrict__ dn,
    float* __restrict__ segP, float* __restrict__ segD) {
  const int tid = threadIdx.x;
  const int s = blockIdx.x >> 2;
  const int b = ((blockIdx.x & 3) << 8) + tid;
  const int t0 = s * SEGLEN;
  float P = 1.f, D = 0.f;
  for (int t = t0 + SEGLEN - 1; t >= t0; --t) {
    const int idx = t * B_DIM + b;
    const float r = rho_ws[idx];
    const float disc = dn[idx] ? 0.f : DISCOUNT_F;
    const float c = fminf(1.f, r);
    const float delta = r * (rw[idx] + disc * vp[idx + B_DIM] - vp[idx]);
    const float a = disc * c;
    D = delta + a * D;
    P *= a;
  }
  segP[s * B_DIM + b] = P;
  segD[s * B_DIM + b] = D;
}

// ---------------------------------------------------------------------------
// Kernel 2b: sequential combine of 16 segment states per column (tiny).
// accIn[s,b] = incoming acc at t = (s+1)*SEGLEN  (vs_minus at segment end).
// ---------------------------------------------------------------------------
__global__ __launch_bounds__(256) void k_comb(
    const float* __restrict__ segP, const float* __restrict__ segD,
    float* __restrict__ accIn) {
  const int b = blockIdx.x * blockDim.x + threadIdx.x;
  float acc = 0.f;                               // vs_minus[T] = 0
#pragma unroll
  for (int s = NSEG - 1; s >= 0; --s) {
    accIn[s * B_DIM + b] = acc;
    acc = segD[s * B_DIM + b] + segP[s * B_DIM + b] * acc;
  }
}

// ---------------------------------------------------------------------------
// Kernel 2c: replay each segment with correct incoming acc; produce partial
// sums of  logp*pg_adv  and  vs_minus^2  per (segment, column).
// vs_{t+1} = acc(at t+1) + value_preds[t+1]  (works at t=T-1: acc=0, v[T]).
// ---------------------------------------------------------------------------
__global__ __launch_bounds__(256) void k_replay(
    const float* __restrict__ rho_ws, const float* __restrict__ logp_ws,
    const float* __restrict__ vp, const float* __restrict__ rw,
    const int* __restrict__ dn, const float* __restrict__ accIn,
    float* __restrict__ partP, float* __restrict__ partV) {
  const int tid = threadIdx.x;
  const int s = blockIdx.x >> 2;
  const int b = ((blockIdx.x & 3) << 8) + tid;
  const int t0 = s * SEGLEN;
  float acc = accIn[s * B_DIM + b];
  float pp = 0.f, pv = 0.f;
  for (int t = t0 + SEGLEN - 1; t >= t0; --t) {
    const int idx = t * B_DIM + b;
    const float r = rho_ws[idx];
    const float disc = dn[idx] ? 0.f : DISCOUNT_F;
    const float c = fminf(1.f, r);
    const float rew = rw[idx];
    const float v_t = vp[idx];
    const float v_tp1 = vp[idx + B_DIM];
    const float vs_tp1 = acc + v_tp1;            // vs[t+1]
    const float delta = r * (rew + disc * v_tp1 - v_t);
    acc = delta + disc * c * acc;                // vs_minus[t]
    const float adv = r * (rew + disc * vs_tp1 - v_t);
    pp += logp_ws[idx] * adv;
    pv += acc * acc;                             // (vs - v)^2
  }
  partP[s * B_DIM + b] = pp;
  partV[s * B_DIM + b] = pv;
}

// ---------------------------------------------------------------------------
// Kernel 3: deterministic final reduction -> scalar loss.
// ---------------------------------------------------------------------------
__global__ __launch_bounds__(256) void k_final(
    const float* __restrict__ entP, const float* __restrict__ partP,
    const float* __restrict__ partV, float* __restrict__ out) {
  __shared__ float sE[256], sP[256], sV[256];
  const int tid = threadIdx.x;
  float e = 0.f, p = 0.f, v = 0.f;
  for (int i = tid; i < NBLK_ROWS; i += 256) e += entP[i];
  for (int i = tid; i < NB; i += 256) { p += partP[i]; v += partV[i]; }
  sE[tid] = e; sP[tid] = p; sV[tid] = v;
  __syncthreads();
#pragma unroll
  for (int w = 128; w > 0; w >>= 1) {
    if (tid < w) { sE[tid] += sE[tid + w]; sP[tid] += sP[tid + w]; sV[tid] += sV[tid + w]; }
    __syncthreads();
  }
  if (tid == 0) {
    const float tb = (float)TB;
    const float loss_p = -sP[0] / tb;
    const float loss_v = 0.5f * sV[0] / tb;
    const float mean_e = sE[0] / (float)ROWS_TOTAL;
    out[0] = loss_p + 0.5f * loss_v - 0.001f * mean_e;
  }
}

// ---------------------------------------------------------------------------
extern "C" void kernel_launch(void* const* d_in, const int* in_sizes, int n_in,
                              void* d_out, int out_size, void* d_ws, size_t ws_size,
                              hipStream_t stream) {
  const float* tlog = (const float*)d_in[0];   // (T+1, B, A)
  const float* blog = (const float*)d_in[1];   // (T+1, B, A)
  const float* vp   = (const float*)d_in[2];   // (T+1, B)
  const float* rw   = (const float*)d_in[3];   // (T, B)
  const int*   act  = (const int*)d_in[4];     // (T+1, B)
  const int*   dn   = (const int*)d_in[5];     // (T, B)
  float* out = (float*)d_out;

  float* ws    = (float*)d_ws;
  float* rho   = ws;                 // TB
  float* logp  = rho + TB;           // TB
  float* segP  = logp + TB;          // NB
  float* segD  = segP + NB;          // NB
  float* accIn = segD + NB;          // NB
  float* partP = accIn + NB;         // NB
  float* partV = partP + NB;         // NB
  float* entP  = partV + NB;         // NBLK_ROWS

  k_rows<<<NBLK_ROWS, ROWS_PER_BLOCK, 0, stream>>>(tlog, blog, act, rho, logp, entP);
  k_seg<<<NSEG * 4, 256, 0, stream>>>(rho, vp, rw, dn, segP, segD);
  k_comb<<<4, 256, 0, stream>>>(segP, segD, accIn);
  k_replay<<<NSEG * 4, 256, 0, stream>>>(rho, logp, vp, rw, dn, accIn, partP, partV);
  k_final<<<1, 256, 0, stream>>>(entP, partP, partV, out);
}